// TIm_27221502722176
// MI455X (gfx1250) — compile-verified
//
#include <hip/hip_runtime.h>
#include <cstdint>

// Problem constants (from setup_inputs): B=2, C=4, G=128, T=96
#define Gn 128
#define Bn 2
#define Cn 4
#define Tn 96
#define G3 (Gn * Gn * Gn)
#define T3 (Tn * Tn * Tn)
#define OUT0 (Bn * Cn * G3)   // elements of output #1 before defp starts

// Native CDNA5 transcendental tanh (V_TANH_F32, TRANS32 class).
// Trailing v_nop satisfies the TRANS data hazard: one op must follow a trans op
// before its sources are overwritten or its result is used (ISA §7.4).
__device__ __forceinline__ float tanh_hw(float x) {
    float r;
    asm volatile("v_tanh_f32 %0, %1\n\tv_nop" : "=v"(r) : "v"(x));
    return r;
}

// Fused: defp transpose + tanh warp + trilinear grid_sample + correction add.
// Tile: fixed (b, h); 32 (d) x 32 (w) spatial tile; 1024 threads = 32 wave32.
__global__ __launch_bounds__(1024, 1) void warp_sample_kernel(
    const float* __restrict__ grids,     // [B, D, H, W, 3]
    const float* __restrict__ deform,    // [B, 3, W, H, D]
    const float* __restrict__ corr,      // [B, C, W, H, D]
    const float* __restrict__ tpl,       // [B, C, T, T, T]
    float* __restrict__ out)             // [OUT0] out  ++ [B,D,H,W,3] defp
{
    // 33-float row pad: stride 33 is coprime with 64 banks -> conflict-free
    // transposed reads in both directions.
    __shared__ float sdef[3][32][33];    // [k][w_local][d_local]
    __shared__ float sfld[3][32][33];    // [k][d_local][w_local]

    const int b  = blockIdx.z;
    const int h  = blockIdx.y;
    const int wt = blockIdx.x >> 2;
    const int dt = blockIdx.x & 3;
    const int d0 = dt * 32;
    const int w0 = wt * 32;

    const int lin = threadIdx.x;
    const int lo  = lin & 31;            // lane within wave
    const int hi  = lin >> 5;            // wave id (0..31)

    // ---------- Issue ALL independent global streams up front (max MLP) ----

    // (1) Async-copy deformation tile into LDS (transpose staging).
    //     deformation idx = ((b*3+k)*G + w)*G*G + h*G + d  (d fastest -> lanes).
    {
        const int d = d0 + lo;
        const int w = w0 + hi;
        const int base = (b * 3) * G3 + w * (Gn * Gn) + h * Gn + d;
        #pragma unroll
        for (int k = 0; k < 3; ++k) {
            const float* gp = deform + base + k * G3;
            unsigned lds = (unsigned)(uintptr_t)&sdef[k][hi][lo]; // [k][w][d]
            asm volatile("global_load_async_to_lds_b32 %0, %1, off th:TH_LOAD_NT"
                         :: "v"(lds), "v"(gp) : "memory");
        }
    }

    // (2) Correction loads for phase C (d fastest over lanes -> coalesced).
    //     Loads cannot sink past the clobbering asm wait below, so these are
    //     in flight during the async copies + barrier.
    const int dC = d0 + lo;
    const int wC = w0 + hi;
    const int obase = (b * Cn) * G3 + wC * (Gn * Gn) + h * Gn + dC;
    float cpre[Cn];
    #pragma unroll
    for (int c = 0; c < Cn; ++c)
        cpre[c] = __builtin_nontemporal_load(corr + obase + c * G3);

    // (3) grids loads for phase B (w fastest over lanes -> coalesced).
    const int wB = w0 + lo;
    const int dB = d0 + hi;
    const int gidx = (((b * Gn + dB) * Gn + h) * Gn + wB) * 3;
    const float gx = __builtin_nontemporal_load(grids + gidx + 0);
    const float gy = __builtin_nontemporal_load(grids + gidx + 1);
    const float gz = __builtin_nontemporal_load(grids + gidx + 2);

    // ---------- Drain async copies, then consume ---------------------------
    asm volatile("s_wait_asynccnt 0" ::: "memory");
    __syncthreads();

    // Phase B: transposed defp read, tanh warp field, emit defp output.
    {
        const float dx = sdef[0][lo][hi];   // stride-33 read, conflict-free
        const float dy = sdef[1][lo][hi];
        const float dz = sdef[2][lo][hi];

        // Output #2: defp (same [B,D,H,W,3] layout as grids), write-once.
        float* dp = out + OUT0 + gidx;
        __builtin_nontemporal_store(dx, dp + 0);
        __builtin_nontemporal_store(dy, dp + 1);
        __builtin_nontemporal_store(dz, dp + 2);

        sfld[0][hi][lo] = tanh_hw(gx + dx); // [k][d][w], w contiguous write
        sfld[1][hi][lo] = tanh_hw(gy + dy);
        sfld[2][hi][lo] = tanh_hw(gz + dz);
    }
    __syncthreads();

    // Phase C: trilinear sample (gathers hit L2-resident template),
    //          add pre-fetched correction, store out (d fastest -> coalesced).
    {
        const float fx = sfld[0][lo][hi];   // stride-33 read, conflict-free
        const float fy = sfld[1][lo][hi];
        const float fz = sfld[2][lo][hi];

        // align_corners=True: [-1,1] -> [0, T-1]
        const float ix = (fx + 1.0f) * (0.5f * (Tn - 1));
        const float iy = (fy + 1.0f) * (0.5f * (Tn - 1));
        const float iz = (fz + 1.0f) * (0.5f * (Tn - 1));
        const float xf = floorf(ix), yf = floorf(iy), zf = floorf(iz);
        const float wx = ix - xf, wy = iy - yf, wz = iz - zf;

        int x0 = min(max((int)xf, 0), Tn - 1), x1 = min(x0 + 1, Tn - 1);
        int y0 = min(max((int)yf, 0), Tn - 1), y1 = min(y0 + 1, Tn - 1);
        int z0 = min(max((int)zf, 0), Tn - 1), z1 = min(z0 + 1, Tn - 1);

        const int zy00 = (z0 * Tn + y0) * Tn;
        const int zy01 = (z0 * Tn + y1) * Tn;
        const int zy10 = (z1 * Tn + y0) * Tn;
        const int zy11 = (z1 * Tn + y1) * Tn;

        const int tb = b * Cn * T3;

        #pragma unroll
        for (int c = 0; c < Cn; ++c) {
            const float* tp = tpl + tb + c * T3;   // temporal loads: stay in L2
            const float v000 = tp[zy00 + x0], v001 = tp[zy00 + x1];
            const float v010 = tp[zy01 + x0], v011 = tp[zy01 + x1];
            const float v100 = tp[zy10 + x0], v101 = tp[zy10 + x1];
            const float v110 = tp[zy11 + x0], v111 = tp[zy11 + x1];

            const float c00 = v000 + wx * (v001 - v000);
            const float c01 = v010 + wx * (v011 - v010);
            const float c10 = v100 + wx * (v101 - v100);
            const float c11 = v110 + wx * (v111 - v110);
            const float c0  = c00 + wy * (c01 - c00);
            const float c1  = c10 + wy * (c11 - c10);
            const float s   = c0 + wz * (c1 - c0);

            __builtin_nontemporal_store(s + cpre[c], out + obase + c * G3);
        }
    }
}

extern "C" void kernel_launch(void* const* d_in, const int* in_sizes, int n_in,
                              void* d_out, int out_size, void* d_ws, size_t ws_size,
                              hipStream_t stream) {
    const float* grids   = (const float*)d_in[0];
    const float* deform  = (const float*)d_in[1];
    const float* corr    = (const float*)d_in[2];
    const float* tpl     = (const float*)d_in[3];
    float*       out     = (float*)d_out;

    // grid.x: 4 (d tiles) * 4 (w tiles); grid.y: h; grid.z: b
    dim3 grid(16, Gn, Bn);
    dim3 block(1024);
    warp_sample_kernel<<<grid, block, 0, stream>>>(grids, deform, corr, tpl, out);
}